// FastAttentionRenderer_83099027243675
// MI455X (gfx1250) — compile-verified
//
#include <hip/hip_runtime.h>
#include <math.h>

// ---------------- problem constants ----------------
#define BB      2
#define NN      32768
#define HH      8
#define DHH     64
#define MFEAT   256
#define QKV_LD  1536           // 3 * H * DH
#define CHUNK_T 256            // tokens per block
#define TILES   16             // 16-token tiles per chunk
#define NCHUNK  128            // NN / CHUNK_T
#define NBH     16             // BB * HH
#define NBLK    2048           // NBH * NCHUNK
#define KEPS    1e-4f
#define LNEPS   1e-5f
#define RATIO   0.0625f        // M^-0.5
#define DNORM   0.35355339059327373f   // DH^-0.25

// ---------------- workspace layout (float offsets) ----------------
#define STAB_OFF 0                       // [16]
#define KSUM_OFF 16                      // [16][256]
#define CTX_OFF  8192                    // [16][256][64]
#define PMAX_OFF (8192 + 262144)         // [2048]
// total ~272K floats (~1.1 MB)

typedef __attribute__((ext_vector_type(2))) float v2f;
typedef __attribute__((ext_vector_type(8))) float v8f;

__device__ __forceinline__ v8f wmma4(v2f a, v2f b, v8f c) {
  // D = A(16x4 f32) * B(4x16 f32) + C(16x16 f32)
  return __builtin_amdgcn_wmma_f32_16x16x4_f32(false, a, false, b, (short)0, c,
                                               false, false);
}

// LayerNorm over 4 input features for a 16-token tile -> xnL[16][4]
__device__ __forceinline__ void build_xn(const float* __restrict__ x,
                                         const float* __restrict__ ln_g,
                                         const float* __restrict__ ln_b,
                                         int b, int t0, int tid,
                                         float (*xnL)[4]) {
  if (tid < 16) {
    const float* xp = x + ((b * NN + t0 + tid) << 2);
    float x0 = xp[0], x1 = xp[1], x2 = xp[2], x3 = xp[3];
    float mu = 0.25f * (x0 + x1 + x2 + x3);
    float d0 = x0 - mu, d1 = x1 - mu, d2 = x2 - mu, d3 = x3 - mu;
    float r = rsqrtf(0.25f * (d0 * d0 + d1 * d1 + d2 * d2 + d3 * d3) + LNEPS);
    xnL[tid][0] = d0 * r * ln_g[0] + ln_b[0];
    xnL[tid][1] = d1 * r * ln_g[1] + ln_b[1];
    xnL[tid][2] = d2 * r * ln_g[2] + ln_b[2];
    xnL[tid][3] = d3 * r * ln_g[3] + ln_b[3];
  }
}

__device__ __forceinline__ float qkv_col(const float (*xnL)[4],
                                         const float* __restrict__ w,
                                         const float* __restrict__ bq,
                                         int t, int col) {
  return bq[col] + xnL[t][0] * w[col] + xnL[t][1] * w[QKV_LD + col] +
         xnL[t][2] * w[2 * QKV_LD + col] + xnL[t][3] * w[3 * QKV_LD + col];
}

// ---------------- kernel 0: init accumulators / output bias ----------------
__global__ __launch_bounds__(256) void k_init(const float* __restrict__ b_out,
                                              float* __restrict__ out,
                                              float* __restrict__ ws) {
  int i = blockIdx.x * 256 + threadIdx.x;       // 262144 threads
  if (i < BB * NN * 4) out[i] = b_out[i & 3];
  if (i < NBH * MFEAT * DHH) ws[CTX_OFF + i] = 0.0f;
  if (i < NBH * MFEAT) ws[KSUM_OFF + i] = 0.0f;
}

// ---------------- kernel 1: per-chunk max of k_dash ----------------
__global__ __launch_bounds__(256) void k_kstab(
    const float* __restrict__ x, const float* __restrict__ ln_g,
    const float* __restrict__ ln_b, const float* __restrict__ w_qkv,
    const float* __restrict__ b_qkv, const float* __restrict__ tau,
    const float* __restrict__ proj, float* __restrict__ ws) {
  const int blk = blockIdx.x;
  const int bh = blk / NCHUNK, chunk = blk % NCHUNK;
  const int b = bh >> 3, h = bh & 7;
  const int tid = threadIdx.x, lane = tid & 31, wid = tid >> 5;
  const int fl = lane & 15, kh = (lane >> 4) << 1;
  const float sk = DNORM * rsqrtf(fmaxf(tau[0], 1e-4f));

  // preload projT operand fragments: wave owns feature tiles {2*wid, 2*wid+1}
  v2f pb[2][16];
#pragma unroll
  for (int f = 0; f < 2; ++f) {
    const float* pr = proj + ((2 * wid + f) * 16 + fl) * DHH + kh;
#pragma unroll
    for (int ks = 0; ks < 16; ++ks) pb[f][ks] = *(const v2f*)(pr + 4 * ks);
  }

  __shared__ float xnL[16][4];
  __shared__ float kL[16 * 68];
  __shared__ float red[256];

  float wmax = -3.0e38f;
  const int tbase = chunk * CHUNK_T;
  for (int tt = 0; tt < TILES; ++tt) {
    const int t0 = tbase + tt * 16;
    __syncthreads();
    build_xn(x, ln_g, ln_b, b, t0, tid, xnL);
    __syncthreads();
    for (int i = tid; i < 16 * 64; i += 256) {
      int t = i >> 6, c = i & 63;
      kL[t * 68 + c] = qkv_col(xnL, w_qkv, b_qkv, t, 512 + h * 64 + c) * sk;
    }
    __syncthreads();
#pragma unroll
    for (int f = 0; f < 2; ++f) {
      v8f acc = {};
#pragma unroll
      for (int ks = 0; ks < 16; ++ks) {
        v2f a;
        a.x = kL[fl * 68 + 4 * ks + kh];
        a.y = kL[fl * 68 + 4 * ks + kh + 1];
        acc = wmma4(a, pb[f][ks], acc);
      }
#pragma unroll
      for (int r = 0; r < 8; ++r) wmax = fmaxf(wmax, acc[r]);
    }
  }
  __syncthreads();
  red[tid] = wmax;
  __syncthreads();
  for (int s = 128; s > 0; s >>= 1) {
    if (tid < s) red[tid] = fmaxf(red[tid], red[tid + s]);
    __syncthreads();
  }
  if (tid == 0) ws[PMAX_OFF + blk] = red[0];
}

// ---------------- kernel 2: reduce partial maxima -> stab_k[16] ----------------
__global__ __launch_bounds__(128) void k_stabred(float* __restrict__ ws) {
  __shared__ float r[128];
  const int bh = blockIdx.x, tid = threadIdx.x;
  r[tid] = ws[PMAX_OFF + bh * NCHUNK + tid];
  __syncthreads();
  for (int s = 64; s > 0; s >>= 1) {
    if (tid < s) r[tid] = fmaxf(r[tid], r[tid + s]);
    __syncthreads();
  }
  if (tid == 0) ws[STAB_OFF + bh] = r[0];
}

// ---------------- kernel 3: kp features, ksum and ctx = kp^T @ v ----------------
__global__ __launch_bounds__(256) void k_ctx(
    const float* __restrict__ x, const float* __restrict__ ln_g,
    const float* __restrict__ ln_b, const float* __restrict__ w_qkv,
    const float* __restrict__ b_qkv, const float* __restrict__ tau,
    const float* __restrict__ proj, float* __restrict__ ws) {
  const int blk = blockIdx.x;
  const int bh = blk / NCHUNK, chunk = blk % NCHUNK;
  const int b = bh >> 3, h = bh & 7;
  const int tid = threadIdx.x, lane = tid & 31, wid = tid >> 5;
  const int fl = lane & 15, kh = (lane >> 4) << 1;
  const int mhi = (lane >> 4) << 3;
  const float sk = DNORM * rsqrtf(fmaxf(tau[0], 1e-4f));
  const float stab = ws[STAB_OFF + bh];

  v2f pb[2][16];
#pragma unroll
  for (int f = 0; f < 2; ++f) {
    const float* pr = proj + ((2 * wid + f) * 16 + fl) * DHH + kh;
#pragma unroll
    for (int ks = 0; ks < 16; ++ks) pb[f][ks] = *(const v2f*)(pr + 4 * ks);
  }

  // ctx accumulators: wave owns (feat tiles {2w,2w+1}) x (4 d-tiles) = 8 tiles
  v8f acc[8];
#pragma unroll
  for (int i = 0; i < 8; ++i) acc[i] = {};

  __shared__ float xnL[16][4];
  __shared__ float kL[16 * 68];
  __shared__ float vL[16 * 68];
  __shared__ float kp[16 * 260];
  __shared__ float diag[16];

  float ksAcc = 0.0f;  // per-thread ksum for feature == tid
  const int tbase = chunk * CHUNK_T;
  for (int tt = 0; tt < TILES; ++tt) {
    const int t0 = tbase + tt * 16;
    __syncthreads();
    build_xn(x, ln_g, ln_b, b, t0, tid, xnL);
    __syncthreads();
    for (int i = tid; i < 16 * 128; i += 256) {
      int t = i >> 7, c = i & 127;
      if (c < 64)
        kL[t * 68 + c] = qkv_col(xnL, w_qkv, b_qkv, t, 512 + h * 64 + c) * sk;
      else
        vL[t * 68 + (c - 64)] =
            qkv_col(xnL, w_qkv, b_qkv, t, 1024 + h * 64 + (c - 64));
    }
    __syncthreads();
    if (tid < 16) {
      float s = 0.0f;
      for (int d = 0; d < 64; ++d) {
        float kv = kL[tid * 68 + d];
        s += kv * kv;
      }
      diag[tid] = 0.5f * s;
    }
    __syncthreads();
    // k_dash = kL @ projT, then kp = ratio*(exp(dash - diag - stab) + eps)
#pragma unroll
    for (int f = 0; f < 2; ++f) {
      v8f dsh = {};
#pragma unroll
      for (int ks = 0; ks < 16; ++ks) {
        v2f a;
        a.x = kL[fl * 68 + 4 * ks + kh];
        a.y = kL[fl * 68 + 4 * ks + kh + 1];
        dsh = wmma4(a, pb[f][ks], dsh);
      }
      const int featc = (2 * wid + f) * 16 + fl;
#pragma unroll
      for (int r = 0; r < 8; ++r) {
        int m = r + mhi;
        kp[m * 260 + featc] = RATIO * (__expf(dsh[r] - diag[m] - stab) + KEPS);
      }
    }
    __syncthreads();
    {
      float s = 0.0f;
      for (int t = 0; t < 16; ++t) s += kp[t * 260 + tid];
      ksAcc += s;
    }
    // ctx += kp^T @ v  (A[mfeat][ktok] = kp[ktok][mfeat])
#pragma unroll
    for (int f = 0; f < 2; ++f) {
      const int mfb = (2 * wid + f) * 16;
#pragma unroll
      for (int dt = 0; dt < 4; ++dt) {
        v8f c = acc[f * 4 + dt];
#pragma unroll
        for (int ks = 0; ks < 4; ++ks) {
          v2f a, bb;
          a.x = kp[(4 * ks + kh) * 260 + mfb + fl];
          a.y = kp[(4 * ks + kh + 1) * 260 + mfb + fl];
          bb.x = vL[(4 * ks + kh) * 68 + dt * 16 + fl];
          bb.y = vL[(4 * ks + kh + 1) * 68 + dt * 16 + fl];
          c = wmma4(a, bb, c);
        }
        acc[f * 4 + dt] = c;
      }
    }
  }
  float* ctx = ws + CTX_OFF + bh * (MFEAT * DHH);
#pragma unroll
  for (int f = 0; f < 2; ++f)
#pragma unroll
    for (int dt = 0; dt < 4; ++dt)
#pragma unroll
      for (int r = 0; r < 8; ++r)
        unsafeAtomicAdd(
            &ctx[((2 * wid + f) * 16 + r + mhi) * 64 + dt * 16 + fl],
            acc[f * 4 + dt][r]);
  unsafeAtomicAdd(&ws[KSUM_OFF + bh * 256 + tid], ksAcc);
}

// ---------------- kernel 4: qp features, out = qp @ ctx, final projection ----------------
__global__ __launch_bounds__(256) void k_out(
    const float* __restrict__ x, const float* __restrict__ ln_g,
    const float* __restrict__ ln_b, const float* __restrict__ w_qkv,
    const float* __restrict__ b_qkv, const float* __restrict__ w_out,
    const float* __restrict__ tau, const float* __restrict__ proj,
    float* __restrict__ out, float* __restrict__ ws) {
  const int blk = blockIdx.x;
  const int bh = blk / NCHUNK, chunk = blk % NCHUNK;
  const int b = bh >> 3, h = bh & 7;
  const int tid = threadIdx.x, lane = tid & 31, wid = tid >> 5;
  const int fl = lane & 15, kh = (lane >> 4) << 1;
  const int mhi = (lane >> 4) << 3;
  const float sk = DNORM * rsqrtf(fmaxf(tau[0], 1e-4f));

  v2f pb[2][16];
#pragma unroll
  for (int f = 0; f < 2; ++f) {
    const float* pr = proj + ((2 * wid + f) * 16 + fl) * DHH + kh;
#pragma unroll
    for (int ks = 0; ks < 16; ++ks) pb[f][ks] = *(const v2f*)(pr + 4 * ks);
  }

  // preload ctx operand frags: wave owns d-tile (wid&3), K half (wid>>2)
  const int dt = wid & 3, khw = wid >> 2;
  const float* ctx = ws + CTX_OFF + bh * (MFEAT * DHH);
  v2f cb[32];
#pragma unroll
  for (int ks = 0; ks < 32; ++ks) {
    int k0 = khw * 128 + 4 * ks;
    v2f t;
    t.x = ctx[(k0 + kh) * 64 + dt * 16 + fl];
    t.y = ctx[(k0 + kh + 1) * 64 + dt * 16 + fl];
    cb[ks] = t;
  }

  __shared__ float xnL[16][4];
  __shared__ float qL[16 * 68];
  __shared__ float dashL[16 * 260];
  __shared__ float ksumL[256];
  __shared__ float red[16][17];
  __shared__ float rowst[16];
  __shared__ float diag[16];
  __shared__ float dinv[16];
  __shared__ float obuf[16 * 68];

  ksumL[tid] = ws[KSUM_OFF + bh * 256 + tid];

  const int tbase = chunk * CHUNK_T;
  for (int tt = 0; tt < TILES; ++tt) {
    const int t0 = tbase + tt * 16;
    __syncthreads();
    build_xn(x, ln_g, ln_b, b, t0, tid, xnL);
    __syncthreads();
    for (int i = tid; i < 16 * 64; i += 256) {
      int t = i >> 6, c = i & 63;
      qL[t * 68 + c] = qkv_col(xnL, w_qkv, b_qkv, t, h * 64 + c) * sk;
    }
    __syncthreads();
    if (tid < 16) {
      float s = 0.0f;
      for (int d = 0; d < 64; ++d) {
        float qv = qL[tid * 68 + d];
        s += qv * qv;
      }
      diag[tid] = 0.5f * s;
    }
    // q_dash = qL @ projT
#pragma unroll
    for (int f = 0; f < 2; ++f) {
      v8f dsh = {};
#pragma unroll
      for (int ks = 0; ks < 16; ++ks) {
        v2f a;
        a.x = qL[fl * 68 + 4 * ks + kh];
        a.y = qL[fl * 68 + 4 * ks + kh + 1];
        dsh = wmma4(a, pb[f][ks], dsh);
      }
      const int featc = (2 * wid + f) * 16 + fl;
#pragma unroll
      for (int r = 0; r < 8; ++r) dashL[(r + mhi) * 260 + featc] = dsh[r];
    }
    __syncthreads();
    {  // per-token row max (partial over 16-feature segment)
      const int t = tid >> 4, seg = tid & 15;
      float m = -3.0e38f;
      for (int j = 0; j < 16; ++j) m = fmaxf(m, dashL[t * 260 + seg * 16 + j]);
      red[t][seg] = m;
    }
    __syncthreads();
    if (tid < 16) {
      float m = red[tid][0];
      for (int j = 1; j < 16; ++j) m = fmaxf(m, red[tid][j]);
      rowst[tid] = m;
    }
    __syncthreads();
    {  // qp = ratio*(exp(dash - diag - rowmax) + eps); denom partial vs ksum
      const int t = tid >> 4, seg = tid & 15;
      const float sub = diag[t] + rowst[t];
      float s = 0.0f;
      for (int j = 0; j < 16; ++j) {
        int idx = t * 260 + seg * 16 + j;
        float qv = RATIO * (__expf(dashL[idx] - sub) + KEPS);
        dashL[idx] = qv;
        s += qv * ksumL[seg * 16 + j];
      }
      red[t][seg] = s;
    }
    __syncthreads();
    if (tid < 16) {
      float s = 0.0f;
      for (int j = 0; j < 16; ++j) s += red[tid][j];
      dinv[tid] = 1.0f / s;
    }
    __syncthreads();
    // out_h = qp @ ctx  (each wave: one d-tile, half the K range)
    v8f oacc = {};
#pragma unroll
    for (int ks = 0; ks < 32; ++ks) {
      v2f a;
      a.x = dashL[fl * 260 + khw * 128 + 4 * ks + kh];
      a.y = dashL[fl * 260 + khw * 128 + 4 * ks + kh + 1];
      oacc = wmma4(a, cb[ks], oacc);
    }
    if (khw == 0) {
#pragma unroll
      for (int r = 0; r < 8; ++r) obuf[(r + mhi) * 68 + dt * 16 + fl] = oacc[r];
    }
    __syncthreads();
    if (khw == 1) {
#pragma unroll
      for (int r = 0; r < 8; ++r)
        obuf[(r + mhi) * 68 + dt * 16 + fl] += oacc[r];
    }
    __syncthreads();
    // final 64 -> 4 projection for this head, accumulate into output
    if (tid < 64) {
      const int t = tid >> 2, o = tid & 3;
      float s = 0.0f;
      for (int d = 0; d < 64; ++d)
        s += obuf[t * 68 + d] * w_out[(h * 64 + d) * 4 + o];
      unsafeAtomicAdd(&out[(b * NN + t0 + t) * 4 + o], s * dinv[t]);
    }
  }
}

extern "C" void kernel_launch(void* const* d_in, const int* in_sizes, int n_in,
                              void* d_out, int out_size, void* d_ws,
                              size_t ws_size, hipStream_t stream) {
  const float* x     = (const float*)d_in[0];
  const float* ln_g  = (const float*)d_in[1];
  const float* ln_b  = (const float*)d_in[2];
  const float* w_qkv = (const float*)d_in[3];
  const float* b_qkv = (const float*)d_in[4];
  const float* w_out = (const float*)d_in[5];
  const float* b_out = (const float*)d_in[6];
  const float* tau   = (const float*)d_in[7];
  const float* proj  = (const float*)d_in[8];
  float* out = (float*)d_out;
  float* ws  = (float*)d_ws;

  k_init<<<1024, 256, 0, stream>>>(b_out, out, ws);
  k_kstab<<<NBLK, 256, 0, stream>>>(x, ln_g, ln_b, w_qkv, b_qkv, tau, proj, ws);
  k_stabred<<<NBH, 128, 0, stream>>>(ws);
  k_ctx<<<NBLK, 256, 0, stream>>>(x, ln_g, ln_b, w_qkv, b_qkv, tau, proj, ws);
  k_out<<<NBLK, 256, 0, stream>>>(x, ln_g, ln_b, w_qkv, b_qkv, w_out, tau, proj,
                                  out, ws);
}